// EmbeddingShard_83296595739551
// MI455X (gfx1250) — compile-verified
//
#include <hip/hip_runtime.h>
#include <cstdint>
#include <cstddef>

#define D_MODEL 4096
#define SHARDS 8
#define BLOCK 256

typedef unsigned int uint32x4 __attribute__((ext_vector_type(4)));
typedef int          int32x4  __attribute__((ext_vector_type(4)));
typedef int          int32x8  __attribute__((ext_vector_type(8)));
typedef float        f32x4    __attribute__((ext_vector_type(4)));

#if __has_builtin(__builtin_amdgcn_tensor_load_to_lds)
#define HAVE_TDM 1
#else
#define HAVE_TDM 0
#endif

// ---------------------------------------------------------------------------
// Kernel 1: bsum[d] = sum over shards of b[s][d]   (128 KB -> 16 KB, trivial)
// ---------------------------------------------------------------------------
__global__ void bias_sum_kernel(const float* __restrict__ b,
                                float* __restrict__ bsum) {
    int i = blockIdx.x * blockDim.x + threadIdx.x;
    if (i < D_MODEL) {
        float s = 0.f;
#pragma unroll
        for (int j = 0; j < SHARDS; ++j) s += b[j * D_MODEL + i];
        bsum[i] = s;
    }
}

// ---------------------------------------------------------------------------
// Kernel 2: one block per token. TDM gathers W[tok] (16 KB) into LDS, then
// 8 waves add bias and stream the row out with non-temporal B128 stores.
// bias_mode: 1 -> bvec is precomputed bsum[D]; 0 -> bvec is b[SHARDS][D].
// ---------------------------------------------------------------------------
__global__ __launch_bounds__(BLOCK)
void gather_rows_kernel(const int* __restrict__ tok,
                        const float* __restrict__ W,
                        const float* __restrict__ bvec,
                        int bias_mode,
                        float* __restrict__ out) {
    // Single static LDS allocation -> placed at LDS byte offset 0.
    __shared__ float row[D_MODEL];

    const int t = blockIdx.x;
    const int token = __builtin_amdgcn_readfirstlane(tok[t]);

    // Never-executed store (vocab ids are >= 0, but the compiler can't prove
    // it). This keeps `row` "written" with an unknown runtime value so the
    // LDS loads below can't be value-forwarded to undef, while leaving the
    // pointer un-escaped so the backend still selects ds_load_b128.
    if (token < 0) row[threadIdx.x] = bvec[threadIdx.x];

#if HAVE_TDM
    if ((threadIdx.x >> 5) == 0) {  // only wave 0 issues the DMA
        const unsigned long long gaddr =
            (unsigned long long)(uintptr_t)(W + (size_t)token * D_MODEL);

        // ---- D# group 0 (128b): count / lds_addr / global_addr / type ----
        uint32x4 g0;
        g0.x = 1u;                                     // count=1 valid descriptor
        g0.y = 0u;                                     // lds_addr = 0 (bytes)
        g0.z = (unsigned)(gaddr & 0xFFFFFFFFu);        // global_addr[31:0]
        g0.w = (unsigned)((gaddr >> 32) & 0x01FFFFFFu) // global_addr[56:32]
             | (2u << 30);                             // type = 2 ("image")

        // ---- D# group 1 (256b): sizes / tile dims / strides ----
        int32x8 g1;
        g1[0] = (int)(2u << 16);                       // data_size=2 -> 4B; wg_mask=0
        g1[1] = (int)(((unsigned)D_MODEL & 0xFFFFu) << 16);    // tensor_dim0[15:0]
        g1[2] = (int)(((unsigned)D_MODEL >> 16) | (1u << 16)); // dim0 hi | tensor_dim1=1
        g1[3] = (int)(((unsigned)D_MODEL) << 16);      // tile_dim0 = 4096
        g1[4] = 0;                                     // tile_dim1/2 unused
        g1[5] = (int)D_MODEL;                          // tensor_dim0_stride = 4096
        g1[6] = 0;
        g1[7] = 0;

        int32x4 z4 = {0, 0, 0, 0};
#if __has_include(<hip/amd_detail/amd_gfx1250_TDM.h>)
        int32x8 z8 = {0, 0, 0, 0, 0, 0, 0, 0};
        __builtin_amdgcn_tensor_load_to_lds(g0, g1, z4, z4, z8, 0);
#else
        __builtin_amdgcn_tensor_load_to_lds(g0, g1, z4, z4, 0);
#endif
        __builtin_amdgcn_s_wait_tensorcnt(0);
    }
    __syncthreads();
#else
    // Fallback path (no TDM builtin): direct vectorized global loads into LDS.
    {
        const f32x4* __restrict__ src = (const f32x4*)(W + (size_t)token * D_MODEL);
        f32x4* dst = (f32x4*)row;
#pragma unroll
        for (int k = 0; k < (D_MODEL / 4) / BLOCK; ++k) {
            int i = threadIdx.x + k * BLOCK;
            dst[i] = src[i];
        }
    }
    __syncthreads();
#endif

    // Ordering belt-and-suspenders: nothing below may move above the barrier.
    asm volatile("" ::: "memory");

    const f32x4* __restrict__ row4 = (const f32x4*)row;  // stays LDS-typed
    f32x4* out4 = (f32x4*)(out + (size_t)t * D_MODEL);

    if (bias_mode) {
        const f32x4* __restrict__ bs4 = (const f32x4*)bvec;
#pragma unroll
        for (int k = 0; k < (D_MODEL / 4) / BLOCK; ++k) {
            int i = threadIdx.x + k * BLOCK;
            f32x4 v = row4[i] + bs4[i];
            __builtin_nontemporal_store(v, out4 + i);
        }
    } else {
        // Sum the 8 shard biases inline (bias table is L2-resident).
#pragma unroll
        for (int k = 0; k < (D_MODEL / 4) / BLOCK; ++k) {
            int i = threadIdx.x + k * BLOCK;
            f32x4 bb = {0.f, 0.f, 0.f, 0.f};
#pragma unroll
            for (int s = 0; s < SHARDS; ++s)
                bb += ((const f32x4*)(bvec + (size_t)s * D_MODEL))[i];
            f32x4 v = row4[i] + bb;
            __builtin_nontemporal_store(v, out4 + i);
        }
    }
}

// ---------------------------------------------------------------------------
extern "C" void kernel_launch(void* const* d_in, const int* in_sizes, int n_in,
                              void* d_out, int out_size, void* d_ws, size_t ws_size,
                              hipStream_t stream) {
    const int*   tok = (const int*)d_in[0];    // x: 4096 tokens
    const float* W   = (const float*)d_in[1];  // [50400, 4096] f32
    const float* b   = (const float*)d_in[2];  // [8, 4096] f32
    float*       out = (float*)d_out;          // [T, 4096] f32

    const int T = in_sizes[0];                 // number of tokens

    if (ws_size >= (size_t)D_MODEL * sizeof(float)) {
        float* bsum = (float*)d_ws;
        bias_sum_kernel<<<(D_MODEL + BLOCK - 1) / BLOCK, BLOCK, 0, stream>>>(b, bsum);
        gather_rows_kernel<<<T, BLOCK, 0, stream>>>(tok, W, bsum, 1, out);
    } else {
        gather_rows_kernel<<<T, BLOCK, 0, stream>>>(tok, W, b, 0, out);
    }
}